// Magnus6th_80762565034567
// MI455X (gfx1250) — compile-verified
//
#include <hip/hip_runtime.h>

// CDNA5 / gfx1250, wave32.
// - V_WMMA_F32_16X16X4_F32 for all 8x8 matmuls (two batch elements packed
//   block-diagonally into one 16x16 tile per wave).
// - GLOBAL_LOAD_ASYNC_TO_LDS_B128 (ASYNCcnt) to DMA the A-pair block into LDS.
// - WMMA post-ops fused in the C/D register layout; Horner keeps the
//   loop-invariant A-operand (Omega) cached in VGPRs.

typedef __attribute__((ext_vector_type(2))) float v2f;
typedef __attribute__((ext_vector_type(8))) float v8f;

#define WAVES_PER_BLOCK 4
#define RAW_F 384              // 2 batch elems * 3 matrices * 64 floats
#define BUF_F 256              // one 16x16 fp32 tile
#define NBUF 7
#define YV_F 16
#define WAVE_F (RAW_F + NBUF * BUF_F + YV_F)

#define SQRT15_F 3.8729833462074168852f

// Operand layouts (wave32), K processed in chunks of 4:
//   A 16x4 : vgpr r, lane l -> (M = l&15,                K = k0 + r + 2*(l>>4))
//   B 4x16 : vgpr r, lane l -> (K = k0 + r + 2*(l>>4),   N = l&15)
//   C 16x16: vgpr r, lane l -> (M = r + 8*(l>>4),        N = l&15)

__device__ __forceinline__ void loadAop(v2f a[4], const float* __restrict__ A,
                                        int lane) {
  const int m = lane & 15, hi = lane >> 4;
#pragma unroll
  for (int q = 0; q < 4; ++q) {
    const int ka = 4 * q + 2 * hi;
    a[q].x = A[m * 16 + ka];
    a[q].y = A[m * 16 + ka + 1];
  }
}

__device__ __forceinline__ v8f mmAop(const v2f a[4],
                                     const float* __restrict__ B, int lane) {
  const int n = lane & 15, hi = lane >> 4;
  v8f c = {};
#pragma unroll
  for (int q = 0; q < 4; ++q) {
    const int ka = 4 * q + 2 * hi;
    v2f b;
    b.x = B[ka * 16 + n];
    b.y = B[(ka + 1) * 16 + n];
    c = __builtin_amdgcn_wmma_f32_16x16x4_f32(false, a[q], false, b, (short)0,
                                              c, false, false);
  }
  return c;
}

__device__ __forceinline__ v8f mm(const float* __restrict__ A,
                                  const float* __restrict__ B, int lane) {
  v2f a[4];
  loadAop(a, A, lane);
  return mmAop(a, B, lane);
}

__device__ __forceinline__ void storeC(float* __restrict__ C, v8f c, int lane) {
  const int n = lane & 15, hi = lane >> 4;
#pragma unroll
  for (int r = 0; r < 8; ++r)
    C[(r + 8 * hi) * 16 + n] = c[r];
}

__global__ __launch_bounds__(WAVES_PER_BLOCK * 32)
void magnus6_expm_kernel(const float* __restrict__ Aglob,
                         const float* __restrict__ hPtr,
                         const float* __restrict__ y0,
                         float* __restrict__ out,
                         int npairs) {
  __shared__ float lds[WAVES_PER_BLOCK * WAVE_F];
  const int lane = threadIdx.x & 31;
  const int wid  = threadIdx.x >> 5;
  const int pair = blockIdx.x * WAVES_PER_BLOCK + wid;
  if (pair >= npairs) return;   // wave-uniform; EXEC stays all-ones for WMMA

  float* raw = lds + wid * WAVE_F;
  float* bse = raw + RAW_F;
  float* a1  = bse + 0 * BUF_F;
  float* a2  = bse + 1 * BUF_F;
  float* a3  = bse + 2 * BUF_F;
  float* P   = bse + 3 * BUF_F;
  float* Q   = bse + 4 * BUF_F;
  float* C1  = bse + 5 * BUF_F;
  float* W   = bse + 6 * BUF_F;
  float* yv  = bse + NBUF * BUF_F;

  const float h = hPtr[0];

  // ---- Async DMA both batch elements' A[3][8][8] (1536 B) into LDS. ------
  {
    const unsigned lbase = (unsigned)(uintptr_t)raw + (unsigned)lane * 16u;
    unsigned long long g =
        (unsigned long long)(uintptr_t)(Aglob + (size_t)pair * 384) +
        (unsigned long long)lane * 16ull;
    asm volatile("global_load_async_to_lds_b128 %0, %1, off"
                 :: "v"(lbase), "v"(g) : "memory");
    asm volatile("global_load_async_to_lds_b128 %0, %1, off offset:512"
                 :: "v"(lbase), "v"(g) : "memory");
    asm volatile("global_load_async_to_lds_b128 %0, %1, off offset:1024"
                 :: "v"(lbase), "v"(g) : "memory");
    asm volatile("s_wait_asynccnt 0x0" ::: "memory");
  }
  __builtin_amdgcn_wave_barrier();

  // ---- alpha1/2/3 as block-diagonal 16x16 tiles (off-diag blocks = 0). ----
  const float s2 = h * (SQRT15_F / 3.0f);
  const float s3 = h * (10.0f / 3.0f);
#pragma unroll
  for (int i = 0; i < 8; ++i) {
    const int e = lane * 8 + i;           // contiguous per lane -> b128 LDS ops
    const int r = e >> 4, c = e & 15;
    float x1 = 0.f, x2 = 0.f, x3 = 0.f;
    const bool lo  = (r < 8) & (c < 8);
    const bool hi2 = (r >= 8) & (c >= 8);
    if (lo | hi2) {
      const int off = (hi2 ? 192 : 0) + (r & 7) * 8 + (c & 7);
      x1 = raw[off];         // A[:,0]
      x2 = raw[off + 64];    // A[:,1]
      x3 = raw[off + 128];   // A[:,2]
    }
    a1[e] = h * x2;
    a2[e] = s2 * (x3 - x1);
    a3[e] = s3 * (x1 - 2.0f * x2 + x3);
  }
  __builtin_amdgcn_wave_barrier();

  // ---- C1 = [a1, a2]   (difference fused in registers) --------------------
  {
    v8f cP = mm(a1, a2, lane);
    v8f cQ = mm(a2, a1, lane);
    storeC(C1, cP - cQ, lane);
  }
  __builtin_amdgcn_wave_barrier();
#pragma unroll
  for (int i = 0; i < 8; ++i) {
    const int e = lane * 8 + i;
    W[e] = 2.0f * a3[e] + C1[e];          // W = 2*alpha3 + C1
  }
  __builtin_amdgcn_wave_barrier();

  // ---- C2 = -1/60 * [a1, W]  (fused; overwrites W) ------------------------
  {
    v8f cP = mm(a1, W, lane);
    v8f cQ = mm(W, a1, lane);
    storeC(W, (cQ - cP) * (1.0f / 60.0f), lane);
  }
  __builtin_amdgcn_wave_barrier();
#pragma unroll
  for (int i = 0; i < 8; ++i) {
    const int e = lane * 8 + i;
    P[e] = -20.0f * a1[e] - a3[e] + C1[e];   // term_in_comm
    Q[e] = a2[e] + W[e];                     // term_with_comm
  }
  __builtin_amdgcn_wave_barrier();

  // ---- Omega = a1 + a3/12 + 1/240 * [P, Q] --------------------------------
  {
    v8f cP = mm(P, Q, lane);
    v8f cQ = mm(Q, P, lane);
    storeC(C1, (cP - cQ) * (1.0f / 240.0f), lane);  // commutator term -> C1
  }
  __builtin_amdgcn_wave_barrier();
#pragma unroll
  for (int i = 0; i < 8; ++i) {
    const int e = lane * 8 + i;
    a2[e] = a1[e] + a3[e] * (1.0f / 12.0f) + C1[e]; // Omega (reuse a2)
  }
  __builtin_amdgcn_wave_barrier();

  // ---- U = expm(Omega): degree-7 Taylor, Horner (||Omega|| ~ 2e-2). -------
  float* Om = a2;
  float* T  = P;   // Horner accumulator
#pragma unroll
  for (int i = 0; i < 8; ++i) {
    const int e = lane * 8 + i;
    T[e] = (((e >> 4) == (e & 15)) ? 1.0f : 0.0f) + Om[e] * (1.0f / 7.0f);
  }
  __builtin_amdgcn_wave_barrier();

  {
    v2f omA[4];
    loadAop(omA, Om, lane);               // loop-invariant A-operand in VGPRs
    const int n = lane & 15, hi = lane >> 4;
#pragma unroll
    for (int k = 6; k >= 1; --k) {
      v8f c = mmAop(omA, T, lane);        // Om @ T
      const float rk = 1.0f / (float)k;
#pragma unroll
      for (int r = 0; r < 8; ++r) {       // T = I + c/k, fused in C/D layout
        const int row = r + 8 * hi;
        T[row * 16 + n] = c[r] * rk + ((row == n) ? 1.0f : 0.0f);
      }
      __builtin_amdgcn_wave_barrier();
    }
  }

  // ---- y_next = U @ y0 (block-diagonal => both batch elems at once). ------
  if (lane < 16) {
    const int b = pair * 2 + (lane >> 3);
    yv[lane] = y0[(size_t)b * 8 + (lane & 7)];
  }
  __builtin_amdgcn_wave_barrier();
  if (lane < 16) {
    float acc = 0.f;
#pragma unroll
    for (int j = 0; j < 16; ++j)
      acc += T[lane * 16 + j] * yv[j];
    const int b = pair * 2 + (lane >> 3);
    out[(size_t)b * 8 + (lane & 7)] = acc;
  }
}

extern "C" void kernel_launch(void* const* d_in, const int* in_sizes, int n_in,
                              void* d_out, int out_size, void* d_ws, size_t ws_size,
                              hipStream_t stream) {
  const float* A  = (const float*)d_in[0];
  const float* h  = (const float*)d_in[1];
  const float* y0 = (const float*)d_in[2];
  float* out = (float*)d_out;
  const int B = in_sizes[0] / 192;      // (B,3,8,8) fp32
  const int npairs = B / 2;             // two batch elems per wave (B is even)
  const int blocks = (npairs + WAVES_PER_BLOCK - 1) / WAVES_PER_BLOCK;
  magnus6_expm_kernel<<<blocks, WAVES_PER_BLOCK * 32, 0, stream>>>(A, h, y0, out,
                                                                  npairs);
}